// MultinomialModule_21363167330437
// MI455X (gfx1250) — compile-verified
//
#include <hip/hip_runtime.h>
#include <stdint.h>

#define N_CAT 1000000
#define N_SMP (1024 * 1024)

typedef float v2f __attribute__((ext_vector_type(2)));
typedef float v8f __attribute__((ext_vector_type(8)));
typedef int v4i __attribute__((vector_size(16)));

// workspace layout (bytes)
#define OFF_CDF     0u         // 1,000,000 f32
#define OFF_COARSE  4000000u   // 1024 f32 (977 real + inf pads)
#define OFF_BSUM    4004096u   // 512 f32
#define OFF_BOFF    4006144u   // 512 f32
#define OFF_TOTAL   4008192u   // 1 f32
#define OFF_ACCUM   4008200u   // 1 u64 (8-byte aligned)

// ---------------- Kernel 1: per-chunk (2048 elem) block sums ----------------
__global__ __launch_bounds__(256) void k_blocksum(const float* __restrict__ x,
                                                  float* __restrict__ bsum) {
  __shared__ float red[256];
  const int t = threadIdx.x, b = blockIdx.x;
  const int base = b * 2048;
  float a = 0.f;
#pragma unroll
  for (int i = 0; i < 8; ++i) {
    int idx = base + t + i * 256;
    a += (idx < N_CAT) ? x[idx] : 0.f;
  }
  red[t] = a;
  __syncthreads();
  for (int s = 128; s > 0; s >>= 1) {
    if (t < s) red[t] += red[t + s];
    __syncthreads();
  }
  if (t == 0) bsum[b] = red[0];
}

// ------ Kernel 2: scan block sums, init accumulator, pad coarse table -------
__global__ __launch_bounds__(512) void k_scan_bsums(const float* __restrict__ bsum,
                                                    float* __restrict__ boff,
                                                    float* __restrict__ totalp,
                                                    unsigned long long* __restrict__ accum,
                                                    float* __restrict__ coarse, int nb) {
  __shared__ float s[512];
  const int t = threadIdx.x;
  s[t] = (t < nb) ? bsum[t] : 0.f;
  __syncthreads();
  for (int off = 1; off < 512; off <<= 1) {
    float v = (t >= off) ? s[t - off] : 0.f;
    __syncthreads();
    s[t] += v;
    __syncthreads();
  }
  if (t < nb) boff[t] = (t == 0) ? 0.f : s[t - 1];
  if (t == 0) {
    *totalp = s[nb - 1];
    *accum = 0ULL;
  }
  int ci = 977 + t;  // pad coarse[977..1023] with +inf (never <= target)
  if (ci < 1024) coarse[ci] = __builtin_inff();
}

// --- Kernel 3: WMMA f32 16x16x4 tile scan (exact f32 prefix sum per tile) ---
// Per wave: tile of 256 f32 as 16x16 (element e = m*16 + n).
// Step 1: R = sum_j A_j x U_j  (U upper-triangular ones) -> row-wise scan.
// Step 2: S = sum_j Lstrict_j x Bt_j + R (Bt rows = row totals) -> full scan.
__global__ __launch_bounds__(256) void k_scan_wmma(const float* __restrict__ x,
                                                   const float* __restrict__ boff,
                                                   float* __restrict__ cdf,
                                                   float* __restrict__ coarse) {
  __shared__ float sTile[8];
  const int t = threadIdx.x, b = blockIdx.x;
  const int w = t >> 5, l = t & 31;
  const int m = l & 15, hi = l >> 4;   // A: M = m; B: N = m; K = v + 2*hi (+4j)
  const bool isHi = (hi != 0);
  const int tileBase = b * 2048 + w * 256;

  v2f a[4];
#pragma unroll
  for (int j = 0; j < 4; ++j)
#pragma unroll
    for (int v = 0; v < 2; ++v) {
      int k = 4 * j + 2 * hi + v;
      int idx = tileBase + m * 16 + k;
      a[j][v] = (idx < N_CAT) ? x[idx] : 0.f;
    }

  v8f R = {0.f, 0.f, 0.f, 0.f, 0.f, 0.f, 0.f, 0.f};
#pragma unroll
  for (int j = 0; j < 4; ++j) {
    v2f bu;
#pragma unroll
    for (int v = 0; v < 2; ++v) {
      int k = 4 * j + 2 * hi + v;
      bu[v] = (k <= m) ? 1.f : 0.f;  // U[k][n] = (k <= n), n == l&15
    }
    R = __builtin_amdgcn_wmma_f32_16x16x4_f32(false, a[j], false, bu,
                                              (short)0, R, false, false);
  }

  // Row totals t[r] = R[r][15] (D layout: vgpr r%8, lane 15 + 16*(r/8)).
  // Per (j,v) lanes 0-15 need t[4j+v], lanes 16-31 need t[4j+v+2]: keep both
  // as compile-time-indexed broadcasts so selection is a single cndmask.
  float tLo[4][2], tHi[4][2];
#pragma unroll
  for (int j = 0; j < 4; ++j)
#pragma unroll
    for (int v = 0; v < 2; ++v) {
      const int rlo = 4 * j + v;       // constant after unroll
      const int rhi = rlo + 2;         // constant after unroll
      tLo[j][v] = __shfl(R[rlo & 7], (rlo >> 3) ? 31 : 15, 32);
      tHi[j][v] = __shfl(R[rhi & 7], (rhi >> 3) ? 31 : 15, 32);
    }

  v8f S = R;
#pragma unroll
  for (int j = 0; j < 4; ++j) {
    v2f al, bt;
#pragma unroll
    for (int v = 0; v < 2; ++v) {
      int k = 4 * j + 2 * hi + v;
      al[v] = (k < m) ? 1.f : 0.f;             // Lstrict[m][k] = (k < m)
      bt[v] = isHi ? tHi[j][v] : tLo[j][v];    // Bt[k][n] = t[k] for all n
    }
    S = __builtin_amdgcn_wmma_f32_16x16x4_f32(false, al, false, bt,
                                              (short)0, S, false, false);
  }

  float tileSum = __shfl(S[7], 31, 32);  // S[15][15]
  if (l == 0) sTile[w] = tileSum;
  __syncthreads();
  float off = boff[b];
  for (int w2 = 0; w2 < w; ++w2) off += sTile[w2];

#pragma unroll
  for (int v = 0; v < 8; ++v) {
    int row = v + 8 * hi;
    int idx = tileBase + row * 16 + m;
    if (idx < N_CAT) {
      float val = S[v] + off;
      cdf[idx] = val;
      if (((idx & 1023) == 1023) || (idx == N_CAT - 1)) coarse[idx >> 10] = val;
    }
  }
}

// ----- Kernel 4: binary search (LDS coarse via async-to-LDS, fine in L2) ----
__global__ __launch_bounds__(256) void k_search(const float* __restrict__ u,
                                                const float* __restrict__ cdf,
                                                const float* __restrict__ coarse,
                                                const float* __restrict__ totalp,
                                                unsigned long long* __restrict__ accum) {
  __shared__ float sc[1024];
  __shared__ unsigned long long rs[256];
  const int t = threadIdx.x;

#if __has_builtin(__builtin_amdgcn_global_load_async_to_lds_b128)
  {
    __attribute__((address_space(1))) v4i* g =
        (__attribute__((address_space(1))) v4i*)(coarse + t * 4);
    __attribute__((address_space(3))) v4i* s3 =
        (__attribute__((address_space(3))) v4i*)(sc + t * 4);
    __builtin_amdgcn_global_load_async_to_lds_b128(g, s3, 0, 0);
#if __has_builtin(__builtin_amdgcn_s_wait_asynccnt)
    __builtin_amdgcn_s_wait_asynccnt(0);
#else
    asm volatile("s_wait_asynccnt 0" ::: "memory");
#endif
  }
#else
#pragma unroll
  for (int i = 0; i < 4; ++i) sc[t + i * 256] = coarse[t + i * 256];
#endif
  __syncthreads();

  const float total = *totalp;
  unsigned long long lsum = 0ULL;
  const int gid = blockIdx.x * 256 + t;
#pragma unroll
  for (int s = 0; s < 4; ++s) {
    int si = gid + s * (1024 * 256);
    float target = u[si] * total;  // cdf/total <= u  <=>  cdf <= u*total
    // coarse: count entries <= target among 1024 LDS values (pads = +inf)
    int gc = 0;
#pragma unroll
    for (int step = 512; step > 0; step >>= 1)
      if (sc[gc + step - 1] <= target) gc += step;
    int start = gc << 10;
    int idx;
    if (start >= N_CAT) {
      idx = N_CAT - 1;
    } else {
      int len = N_CAT - start;
      if (len > 1024) len = 1024;
      int c = 0;
#pragma unroll
      for (int step = 512; step > 0; step >>= 1) {
        int p = c + step;
        if (p <= len && cdf[start + p - 1] <= target) c = p;
      }
      idx = start + c;
      if (idx > N_CAT - 1) idx = N_CAT - 1;
    }
    lsum += (unsigned long long)idx;
  }
  rs[t] = lsum;
  __syncthreads();
  for (int s = 128; s > 0; s >>= 1) {
    if (t < s) rs[t] += rs[t + s];
    __syncthreads();
  }
  if (t == 0) atomicAdd(accum, rs[0]);
}

// ----------------------- Kernel 5: finalize the mean ------------------------
__global__ void k_finalize(const unsigned long long* __restrict__ accum,
                           float* __restrict__ out) {
  if (threadIdx.x == 0 && blockIdx.x == 0)
    out[0] = (float)((double)(*accum) * (1.0 / (double)N_SMP));
}

extern "C" void kernel_launch(void* const* d_in, const int* in_sizes, int n_in,
                              void* d_out, int out_size, void* d_ws, size_t ws_size,
                              hipStream_t stream) {
  const float* x = (const float*)d_in[0];
  const float* u = (const float*)d_in[1];
  float* out = (float*)d_out;
  char* ws = (char*)d_ws;
  float* cdf = (float*)(ws + OFF_CDF);
  float* coarse = (float*)(ws + OFF_COARSE);
  float* bsum = (float*)(ws + OFF_BSUM);
  float* boff = (float*)(ws + OFF_BOFF);
  float* totalp = (float*)(ws + OFF_TOTAL);
  unsigned long long* accum = (unsigned long long*)(ws + OFF_ACCUM);
  const int nb = (N_CAT + 2047) / 2048;  // 489

  k_blocksum<<<nb, 256, 0, stream>>>(x, bsum);
  k_scan_bsums<<<1, 512, 0, stream>>>(bsum, boff, totalp, accum, coarse, nb);
  k_scan_wmma<<<nb, 256, 0, stream>>>(x, boff, cdf, coarse);
  k_search<<<1024, 256, 0, stream>>>(u, cdf, coarse, totalp, accum);
  k_finalize<<<1, 1, 0, stream>>>(accum, out);
}